// GeodesicPrototypeClassifier_24043226923959
// MI455X (gfx1250) — compile-verified
//
#include <hip/hip_runtime.h>

typedef float v2f __attribute__((ext_vector_type(2)));
typedef float v8f __attribute__((ext_vector_type(8)));

// Problem constants (from reference)
#define C_CURV      1.0f
#define BALL_EPS    0.001f
#define ATANH_CLAMP (1.0f - 1e-05f)
#define MIN_DENOM   1e-15f
#define DDIM        64          // feature dimension
#define WAVES_PER_BLOCK 8
#define ROWS_PER_BLOCK (WAVES_PER_BLOCK * 16)   // 128 rows of x per block

// One wave computes a 16x16 tile of out = f(X @ P^T) using V_WMMA_F32_16X16X4_F32,
// accumulating K=64 in 16 steps of k=4. Prototypes for the block's 16-column tile
// are ball-projected once into LDS and shared by all 8 waves.
__global__ __launch_bounds__(256)
void geo_proto_kernel(const float* __restrict__ X,
                      const float* __restrict__ P,
                      float* __restrict__ Out,
                      int N, int K)
{
    __shared__ float Pt[16][66];          // projected prototype tile [proto][dim], padded
    __shared__ float p2s[16];             // ||p||^2 per prototype (post-projection)
    __shared__ float x2s[WAVES_PER_BLOCK][16];  // ||x||^2 per row, per wave

    const int tid  = threadIdx.x;
    const int wave = tid >> 5;
    const int lane = tid & 31;
    const int m    = lane & 15;           // A-row index / B-col index / C N-index
    const int h    = lane >> 4;           // half-wave selector (K pair / C M-offset)

    const int col0 = blockIdx.y * 16;                      // prototype tile base
    const int row0 = blockIdx.x * ROWS_PER_BLOCK + wave * 16;  // x-row tile base

    // ---- Phase 1: project 16 prototypes into LDS (one thread per prototype) ----
    if (tid < 16) {
        const float* prow = P + (size_t)(col0 + tid) * DDIM;
        float ss = 0.0f;
        for (int d = 0; d < DDIM; ++d) { float v = prow[d]; ss += v * v; }
        float nrm   = fmaxf(sqrtf(ss), 1e-15f);
        float maxn  = (1.0f - BALL_EPS) / sqrtf(C_CURV);
        float scale = fminf(1.0f, maxn / nrm);
        for (int d = 0; d < DDIM; ++d) Pt[tid][d] = prow[d] * scale;
        p2s[tid] = ss * scale * scale;
    }
    __syncthreads();

    // ---- Phase 2: load A fragments (global) and B fragments (LDS) ----
    // A layout (32-bit, 16x4): lane l holds A[m = l%16][k = 2*(l/16) + {0,1}]
    v2f a[16], b[16];
    const float* xrow = X + (size_t)(row0 + m) * DDIM + 2 * h;
    float part = 0.0f;                    // partial ||x_row||^2 over this lane's dims
#pragma unroll
    for (int kk = 0; kk < 16; ++kk) {
        a[kk] = *(const v2f*)(xrow + 4 * kk);
        b[kk] = *(const v2f*)(&Pt[m][4 * kk + 2 * h]);
        part += a[kk].x * a[kk].x + a[kk].y * a[kk].y;
    }

    // ---- Phase 3: 16 chained fp32 WMMAs: acc(16x16) += A(16x4) x B(4x16) ----
    v8f acc = {};
#pragma unroll
    for (int kk = 0; kk < 16; ++kk) {
        acc = __builtin_amdgcn_wmma_f32_16x16x4_f32(
            /*neg_a=*/false, a[kk],
            /*neg_b=*/false, b[kk],
            /*c_mod=*/(short)0, acc,
            /*reuse_a=*/false, /*reuse_b=*/false);
    }

    // ---- Phase 4: finish x2 per row (lane l and l^16 hold complementary halves) ----
    float x2full = part + __shfl_xor(part, 16, 32);
    if (lane < 16) x2s[wave][lane] = x2full;
    __syncthreads();

    // ---- Phase 5: epilogue. C layout: VGPR r -> M = r + 8*h, N = m ----
    const float p2     = p2s[m];
    const float sqrt_c = sqrtf(C_CURV);
#pragma unroll
    for (int r = 0; r < 8; ++r) {
        const int   row = r + 8 * h;
        const float x2  = x2s[wave][row];
        const float xy  = acc[r];

        const float t  = 1.0f - 2.0f * C_CURV * xy;
        const float A1 = t + C_CURV * p2;              // 1 - 2c*xy + c*p2
        const float B1 = 1.0f - C_CURV * x2;           // 1 - c*x2
        float den = t + C_CURV * C_CURV * x2 * p2;     // 1 - 2c*xy + c^2*x2*p2
        den = fmaxf(den, MIN_DENOM);

        // ||(-A1*x + B1*p)||^2 = A1^2*x2 - 2*A1*B1*xy + B1^2*p2
        float ma2 = (A1 * A1 * x2 - 2.0f * A1 * B1 * xy + B1 * B1 * p2) / (den * den);
        ma2 = fmaxf(ma2, 0.0f);

        float z = sqrt_c * sqrtf(ma2);
        z = fminf(z, ATANH_CLAMP);
        const float at   = 0.5f * __logf((1.0f + z) / (1.0f - z));  // atanh(z)
        const float dist = (2.0f / sqrt_c) * at;

        Out[(size_t)(row0 + row) * K + (col0 + m)] = -(dist * dist);
    }
}

extern "C" void kernel_launch(void* const* d_in, const int* in_sizes, int n_in,
                              void* d_out, int out_size, void* d_ws, size_t ws_size,
                              hipStream_t stream) {
    const float* x = (const float*)d_in[0];   // (N, 64) fp32
    const float* p = (const float*)d_in[1];   // (K, 64) fp32
    float* out = (float*)d_out;               // (N, K) fp32

    const int N = in_sizes[0] / DDIM;         // 16384
    const int K = in_sizes[1] / DDIM;         // 128

    dim3 grid(N / ROWS_PER_BLOCK, K / 16);    // (128, 8)
    geo_proto_kernel<<<grid, 256, 0, stream>>>(x, p, out, N, K);
}